// ReductionAELossWoClassify_16578573763403
// MI455X (gfx1250) — compile-verified
//
#include <hip/hip_runtime.h>

// ReductionAELossWoClassify: out = 1 - mean_n( sum_d A[n,d]*B[n,d] )   (l2_loss == 0)
//                            out = mean( (A-B)^2 )                      (l2_loss != 0)
// A,B: f32 [8192, 2048]. Memory-bound streaming reduction:
//   134 MB / 23.3 TB/s ~= 5.8 us floor. Direct-to-VGPR b128 NT loads,
//   4x-unrolled grid-stride (8 independent b128 loads in flight per wave),
//   deterministic two-pass tree reduction (no atomics).
// CDNA5 path: v_wmma_f32_16x16x4_f32 for the wave32 reduction epilogue.

typedef float v2f __attribute__((ext_vector_type(2)));
typedef float v4f __attribute__((ext_vector_type(4)));
typedef float v8f __attribute__((ext_vector_type(8)));

// Sum x over all 32 lanes of the wave using one f32 WMMA.
//   A(16x4) = per-lane data (a0 = x, a1 = 0), B(4x16) = ones, C = 0.
//   D[m,n] = rowsum_m(A); per-lane sum of the 8 D VGPRs = sum of rowsums
//   {0..7} (lanes 0-15) or {8..15} (lanes 16-31); one cross-half-wave add
//   yields the total of all 64 A entries = sum_lanes(x). Layout-independent.
// Requires EXEC == all ones (call sites keep control flow uniform).
__device__ __forceinline__ float wave_sum32(float x) {
  v2f a;
  a[0] = x;
  a[1] = 0.0f;
  v2f b;
  b[0] = 1.0f;
  b[1] = 1.0f;
  v8f c = {};
  v8f d = __builtin_amdgcn_wmma_f32_16x16x4_f32(
      /*neg_a=*/false, a, /*neg_b=*/false, b,
      /*c_mod=*/(short)0, c, /*reuse_a=*/false, /*reuse_b=*/false);
  float s = ((d[0] + d[1]) + (d[2] + d[3])) + ((d[4] + d[5]) + (d[6] + d[7]));
  s += __shfl_down(s, 16, 32);  // lanes 0..15 now hold the full 32-lane sum
  return s;
}

__device__ __forceinline__ float block_sum(float acc) {
  __shared__ float smem[8];
  const int lane = threadIdx.x & 31;
  const int wv = threadIdx.x >> 5;
  const float w = wave_sum32(acc);
  if (lane == 0) smem[wv] = w;
  __syncthreads();
  float s = 0.0f;
  if (threadIdx.x == 0) {
#pragma unroll
    for (int j = 0; j < 8; ++j) s += smem[j];
  }
  return s;  // valid on thread 0
}

__global__ void __launch_bounds__(256)
partial_kernel(const v4f* __restrict__ A, const v4f* __restrict__ B,
               const int* __restrict__ l2flag, float* __restrict__ partials,
               int total4) {
  const int flag = *l2flag;  // uniform scalar load
  const int stride = (int)(gridDim.x * blockDim.x);
  const int tid = (int)(blockIdx.x * blockDim.x + threadIdx.x);

  float acc0 = 0.0f, acc1 = 0.0f;  // two chains -> shorter FMA dependency
  int i = tid;

  if (flag == 0) {
    // 4x unroll: 8 independent b128 loads in flight before first consumer.
    for (; i + 3 * stride < total4; i += 4 * stride) {
      const v4f a0 = __builtin_nontemporal_load(&A[i]);
      const v4f a1 = __builtin_nontemporal_load(&A[i + stride]);
      const v4f a2 = __builtin_nontemporal_load(&A[i + 2 * stride]);
      const v4f a3 = __builtin_nontemporal_load(&A[i + 3 * stride]);
      const v4f b0 = __builtin_nontemporal_load(&B[i]);
      const v4f b1 = __builtin_nontemporal_load(&B[i + stride]);
      const v4f b2 = __builtin_nontemporal_load(&B[i + 2 * stride]);
      const v4f b3 = __builtin_nontemporal_load(&B[i + 3 * stride]);
      acc0 = fmaf(a0[0], b0[0], acc0);
      acc1 = fmaf(a0[1], b0[1], acc1);
      acc0 = fmaf(a0[2], b0[2], acc0);
      acc1 = fmaf(a0[3], b0[3], acc1);
      acc0 = fmaf(a1[0], b1[0], acc0);
      acc1 = fmaf(a1[1], b1[1], acc1);
      acc0 = fmaf(a1[2], b1[2], acc0);
      acc1 = fmaf(a1[3], b1[3], acc1);
      acc0 = fmaf(a2[0], b2[0], acc0);
      acc1 = fmaf(a2[1], b2[1], acc1);
      acc0 = fmaf(a2[2], b2[2], acc0);
      acc1 = fmaf(a2[3], b2[3], acc1);
      acc0 = fmaf(a3[0], b3[0], acc0);
      acc1 = fmaf(a3[1], b3[1], acc1);
      acc0 = fmaf(a3[2], b3[2], acc0);
      acc1 = fmaf(a3[3], b3[3], acc1);
    }
    for (; i < total4; i += stride) {  // tail (empty for the reference shape)
      const v4f a = __builtin_nontemporal_load(&A[i]);
      const v4f b = __builtin_nontemporal_load(&B[i]);
      acc0 = fmaf(a[0], b[0], acc0);
      acc1 = fmaf(a[1], b[1], acc1);
      acc0 = fmaf(a[2], b[2], acc0);
      acc1 = fmaf(a[3], b[3], acc1);
    }
  } else {
    for (; i + 3 * stride < total4; i += 4 * stride) {
      const v4f a0 = __builtin_nontemporal_load(&A[i]);
      const v4f a1 = __builtin_nontemporal_load(&A[i + stride]);
      const v4f a2 = __builtin_nontemporal_load(&A[i + 2 * stride]);
      const v4f a3 = __builtin_nontemporal_load(&A[i + 3 * stride]);
      const v4f b0 = __builtin_nontemporal_load(&B[i]);
      const v4f b1 = __builtin_nontemporal_load(&B[i + stride]);
      const v4f b2 = __builtin_nontemporal_load(&B[i + 2 * stride]);
      const v4f b3 = __builtin_nontemporal_load(&B[i + 3 * stride]);
      const v4f d0 = a0 - b0;
      const v4f d1 = a1 - b1;
      const v4f d2 = a2 - b2;
      const v4f d3 = a3 - b3;
      acc0 = fmaf(d0[0], d0[0], acc0);
      acc1 = fmaf(d0[1], d0[1], acc1);
      acc0 = fmaf(d0[2], d0[2], acc0);
      acc1 = fmaf(d0[3], d0[3], acc1);
      acc0 = fmaf(d1[0], d1[0], acc0);
      acc1 = fmaf(d1[1], d1[1], acc1);
      acc0 = fmaf(d1[2], d1[2], acc0);
      acc1 = fmaf(d1[3], d1[3], acc1);
      acc0 = fmaf(d2[0], d2[0], acc0);
      acc1 = fmaf(d2[1], d2[1], acc1);
      acc0 = fmaf(d2[2], d2[2], acc0);
      acc1 = fmaf(d2[3], d2[3], acc1);
      acc0 = fmaf(d3[0], d3[0], acc0);
      acc1 = fmaf(d3[1], d3[1], acc1);
      acc0 = fmaf(d3[2], d3[2], acc0);
      acc1 = fmaf(d3[3], d3[3], acc1);
    }
    for (; i < total4; i += stride) {
      const v4f a = __builtin_nontemporal_load(&A[i]);
      const v4f b = __builtin_nontemporal_load(&B[i]);
      const v4f d = a - b;
      acc0 = fmaf(d[0], d[0], acc0);
      acc1 = fmaf(d[1], d[1], acc1);
      acc0 = fmaf(d[2], d[2], acc0);
      acc1 = fmaf(d[3], d[3], acc1);
    }
  }

  const float s = block_sum(acc0 + acc1);
  if (threadIdx.x == 0) partials[blockIdx.x] = s;
}

__global__ void __launch_bounds__(256)
finalize_kernel(const float* __restrict__ partials,
                const int* __restrict__ l2flag, float* __restrict__ out,
                int nPartials, float nRows, float nElems) {
  float acc = 0.0f;
  for (int i = (int)threadIdx.x; i < nPartials; i += (int)blockDim.x)
    acc += partials[i];

  const float s = block_sum(acc);
  if (threadIdx.x == 0) {
    if (*l2flag != 0)
      out[0] = s / nElems;        // mean((A-B)^2)
    else
      out[0] = 1.0f - s / nRows;  // 1 - mean(rowwise dot)
  }
}

extern "C" void kernel_launch(void* const* d_in, const int* in_sizes, int n_in,
                              void* d_out, int out_size, void* d_ws,
                              size_t ws_size, hipStream_t stream) {
  const v4f* A = (const v4f*)d_in[0];
  const v4f* B = (const v4f*)d_in[1];
  const int* l2flag = (const int*)d_in[2];
  float* out = (float*)d_out;
  float* partials = (float*)d_ws;  // needs nBlocks * 4 bytes

  const int totalElems = in_sizes[0];      // N * D = 16,777,216
  const int total4 = totalElems >> 2;      // 4,194,304 float4 pairs
  const int D = 2048;
  const float nRows = (float)(totalElems / D);  // 8192
  const float nElems = (float)totalElems;

  const int nBlocks = 1024;  // 8 wave32s/block; 4 unrolled iters/thread

  partial_kernel<<<nBlocks, 256, 0, stream>>>(A, B, l2flag, partials, total4);
  finalize_kernel<<<1, 256, 0, stream>>>(partials, l2flag, out, nBlocks, nRows,
                                         nElems);
}